// SelfAttention_4045859192939
// MI455X (gfx1250) — compile-verified
//
#include <hip/hip_runtime.h>
#include <hip/hip_bf16.h>

typedef __attribute__((ext_vector_type(2))) float v2f;
typedef __attribute__((ext_vector_type(8))) float v8f;

#define N_ATOMS 8192
#define N_PAIR  131072
#define N_DEG   3
#define N_HEAD  8
#define F_DIM   128
#define D_HEAD  16
#define K_RBF   32
#define H_FILT  64
#define M_TOT   9
#define OUTC    (N_DEG * F_DIM)   // 384
#define H1S     68                // padded LDS row stride for h1 (bank-conflict-free)
#define LOG2F_  0.69314718055994530942f

static __device__ __forceinline__ v8f wmma_f32_16x16x4(v2f a, v2f b, v8f c) {
  // 8 args: (neg_a, A, neg_b, B, c_mod, C, reuse_a, reuse_b)
  return __builtin_amdgcn_wmma_f32_16x16x4_f32(false, a, false, b, (short)0, c, false, false);
}

// ---------------------------------------------------------------------------
// Zero the output (scatter kernel accumulates with atomics).
// ---------------------------------------------------------------------------
__global__ void zero_kernel(float4* __restrict__ out, int n4) {
  int i = blockIdx.x * blockDim.x + threadIdx.x;
  if (i < n4) out[i] = make_float4(0.f, 0.f, 0.f, 0.f);
}

// ---------------------------------------------------------------------------
// Kernel A: per-atom x_inv = sum of squares over spherical repeats, then
// Q = x_inv @ W_Q^T and K = x_inv @ W_K^T per (degree, head) via
// V_WMMA_F32_16X16X4_F32 on 16-atom tiles. One wave per head, 8 waves/block.
// ---------------------------------------------------------------------------
__global__ __launch_bounds__(256)
void qk_project_kernel(const float* __restrict__ x,
                       const float* __restrict__ W_Q,
                       const float* __restrict__ W_K,
                       float* __restrict__ Qws,
                       float* __restrict__ Kws) {
  const int lane  = threadIdx.x & 31;
  const int head  = threadIdx.x >> 5;   // wave id == head (8 waves)
  const int lh    = lane >> 4;          // lane half (0/1)
  const int lrow  = lane & 15;
  const int atom0 = blockIdx.x * 16;

  const int seg_start[3] = {0, 1, 4};
  const int seg_cnt[3]   = {1, 3, 5};

  for (int g = 0; g < N_DEG; ++g) {
    // ---- A fragments: x_inv tile (16 atoms x 16 j), K split into 4 steps of 4
    v2f afrag[4];
    {
      const int atom = atom0 + lrow;
      const float* xp = x + (size_t)atom * M_TOT * F_DIM + head * D_HEAD;
      for (int k = 0; k < 4; ++k) {
        const int j0 = 4 * k + 2 * lh;
        float s0 = 0.f, s1 = 0.f;
        for (int r = 0; r < seg_cnt[g]; ++r) {
          const int m = seg_start[g] + r;
          float a0 = xp[m * F_DIM + j0];
          float a1 = xp[m * F_DIM + j0 + 1];
          s0 = fmaf(a0, a0, s0);
          s1 = fmaf(a1, a1, s1);
        }
        afrag[k].x = s0;
        afrag[k].y = s1;
      }
    }
    // ---- Q tile: B[j][i] = W_Q[g,h,i,j]  (N = i = lrow, K = j)
    {
      const float* wq = W_Q + ((size_t)(g * N_HEAD + head) * D_HEAD + lrow) * D_HEAD;
      v8f acc = {};
      for (int k = 0; k < 4; ++k) {
        const int j0 = 4 * k + 2 * lh;
        v2f b;
        b.x = wq[j0];
        b.y = wq[j0 + 1];
        acc = wmma_f32_16x16x4(afrag[k], b, acc);
      }
      for (int r = 0; r < 8; ++r) {
        const int row = r + 8 * lh;
        Qws[((size_t)(atom0 + row) * N_DEG + g) * F_DIM + head * D_HEAD + lrow] = acc[r];
      }
    }
    // ---- K tile
    {
      const float* wk = W_K + ((size_t)(g * N_HEAD + head) * D_HEAD + lrow) * D_HEAD;
      v8f acc = {};
      for (int k = 0; k < 4; ++k) {
        const int j0 = 4 * k + 2 * lh;
        v2f b;
        b.x = wk[j0];
        b.y = wk[j0 + 1];
        acc = wmma_f32_16x16x4(afrag[k], b, acc);
      }
      for (int r = 0; r < 8; ++r) {
        const int row = r + 8 * lh;
        Kws[((size_t)(atom0 + row) * N_DEG + g) * F_DIM + head * D_HEAD + lrow] = acc[r];
      }
    }
  }
}

// ---------------------------------------------------------------------------
// Kernel B: per-pair MLP (two WMMA GEMM layers), fused alpha = sum_d Qi*Kj*Wij
// reduction, and segment-accumulated atomic scatter (idx_i is sorted, so
// consecutive pairs usually share a destination atom -> accumulate in regs,
// flush once per segment).
// 8 waves/block, 16 pairs/wave -> 128 pairs/block.
// ---------------------------------------------------------------------------
__global__ __launch_bounds__(256)
void pair_kernel(const float* __restrict__ rbf,
                 const float* __restrict__ phi_r_cut,
                 const float* __restrict__ mlp_W1,
                 const float* __restrict__ mlp_b1,
                 const float* __restrict__ mlp_W2,
                 const float* __restrict__ mlp_b2,
                 const int*   __restrict__ idx_i,
                 const int*   __restrict__ idx_j,
                 const float* __restrict__ pair_mask,
                 const float* __restrict__ Qws,
                 const float* __restrict__ Kws,
                 const float* __restrict__ x,
                 float* __restrict__ out) {
  __shared__ float s_W1[K_RBF * H_FILT];        //  8 KB
  __shared__ float s_h1[8][16 * H1S];           // ~34 KB (padded rows)
  __shared__ float s_alpha[8][16 * 24];         // 12 KB

  const int tid = threadIdx.x;
  // warm L2 with W2 (96 KB, reused by every block) -> global_prefetch_b8
  for (int i = tid * 32; i < H_FILT * OUTC; i += 256 * 32)
    __builtin_prefetch(&mlp_W2[i], 0, 1);
  for (int i = tid; i < K_RBF * H_FILT; i += 256) s_W1[i] = mlp_W1[i];
  __syncthreads();

  const int lane = tid & 31;
  const int w    = tid >> 5;
  const int lh   = lane >> 4;
  const int lrow = lane & 15;
  const int p0   = blockIdx.x * 128 + w * 16;

  // ---- layer 1: H1(16x64) = RBF(16x32) @ W1(32x64), K in 8 steps of 4
  v2f a1[8];
  {
    const float* rp = rbf + (size_t)(p0 + lrow) * K_RBF;
    for (int k = 0; k < 8; ++k) {
      const int j0 = 4 * k + 2 * lh;
      a1[k] = *(const v2f*)(rp + j0);
    }
  }
  for (int t = 0; t < 4; ++t) {
    v8f acc = {};
    for (int k = 0; k < 8; ++k) {
      const int j0 = 4 * k + 2 * lh;
      v2f b;
      b.x = s_W1[j0 * H_FILT + t * 16 + lrow];
      b.y = s_W1[(j0 + 1) * H_FILT + t * 16 + lrow];
      acc = wmma_f32_16x16x4(a1[k], b, acc);
    }
    const float bb = mlp_b1[t * 16 + lrow];
    for (int r = 0; r < 8; ++r) {
      float v  = acc[r] + bb;
      // shifted softplus: log(1+e^v) - log 2, numerically stable
      float sp = fmaxf(v, 0.f) + __logf(1.f + __expf(-fabsf(v))) - LOG2F_;
      s_h1[w][(r + 8 * lh) * H1S + t * 16 + lrow] = sp;
    }
  }
  __syncthreads();

  // ---- per-pair scalars for this lane's accumulator rows
  int   ii[8], jj[8];
  float ph[8], mk2[8];
  for (int r = 0; r < 8; ++r) {
    const int pr = p0 + r + 8 * lh;
    ii[r] = idx_i[pr];
    jj[r] = idx_j[pr];
    ph[r] = phi_r_cut[pr];
    const float m = pair_mask[pr];
    mk2[r] = m * m;   // mask hits both Q_i and K_j in the reference
  }

  // ---- h1 A-fragments (K = 64 -> 16 steps of 4), read back from LDS
  v2f a2[16];
  for (int k = 0; k < 16; ++k) {
    const int j0 = 4 * k + 2 * lh;
    a2[k] = *(const v2f*)&s_h1[w][lrow * H1S + j0];
  }

  // ---- layer 2 tiles: each 16-col tile == one (deg g, head h) slice.
  //      Fuse W_ij with Q_i*K_j elementwise + reduce over d -> alpha.
  for (int t = 0; t < 24; ++t) {
    const int g = t >> 3;
    const int h = t & 7;
    v8f acc = {};
    for (int k = 0; k < 16; ++k) {
      const int j0 = 4 * k + 2 * lh;
      v2f b;
      b.x = mlp_W2[j0 * OUTC + t * 16 + lrow];        // W2 lives in L2
      b.y = mlp_W2[(j0 + 1) * OUTC + t * 16 + lrow];
      acc = wmma_f32_16x16x4(a2[k], b, acc);
    }
    const float b2v = mlp_b2[t * 16 + lrow];
    for (int r = 0; r < 8; ++r) {
      const float wij = (acc[r] + b2v) * ph[r];
      const float qi = Qws[((size_t)ii[r] * N_DEG + g) * F_DIM + h * D_HEAD + lrow];
      const float kj = Kws[((size_t)jj[r] * N_DEG + g) * F_DIM + h * D_HEAD + lrow];
      float v = wij * qi * kj * mk2[r];
      // reduce over d (16 lanes of this half); masks < 16 never cross halves
      v += __shfl_xor(v, 8);
      v += __shfl_xor(v, 4);
      v += __shfl_xor(v, 2);
      v += __shfl_xor(v, 1);
      if (lrow == 0)
        s_alpha[w][(r + 8 * lh) * 24 + t] = v * 0.25f;   // 1/sqrt(D_HEAD)
    }
  }
  __syncthreads();

  // ---- scatter with segment accumulation (idx_i is sorted).
  //      Each lane owns f = q*32 + lane for q=0..3, all 9 m-channels:
  //      36 running sums; flush to global atomics only when idx_i changes.
  {
    float accz[M_TOT * 4];
#pragma unroll
    for (int s = 0; s < M_TOT * 4; ++s) accz[s] = 0.f;

    int cur_i = idx_i[p0];
    for (int p = 0; p < 16; ++p) {
      const int pr = p0 + p;
      const int i  = idx_i[pr];
      if (i != cur_i) {
        float* outp = out + (size_t)cur_i * M_TOT * F_DIM;
#pragma unroll
        for (int mm = 0; mm < M_TOT; ++mm)
#pragma unroll
          for (int q = 0; q < 4; ++q) {
            unsafeAtomicAdd(&outp[mm * F_DIM + q * 32 + lane], accz[mm * 4 + q]);
            accz[mm * 4 + q] = 0.f;
          }
        cur_i = i;
      }
      const int   j = idx_j[pr];
      const float m = pair_mask[pr];
      const float* xj = x + (size_t)j * M_TOT * F_DIM;
#pragma unroll
      for (int mm = 0; mm < M_TOT; ++mm) {
        const int g = (mm == 0) ? 0 : ((mm < 4) ? 1 : 2);
#pragma unroll
        for (int q = 0; q < 4; ++q) {
          const int f = q * 32 + lane;
          const float a = s_alpha[w][p * 24 + g * 8 + (q * 2 + lh)];
          accz[mm * 4 + q] = fmaf(a * m, xj[mm * F_DIM + f], accz[mm * 4 + q]);
        }
      }
    }
    // final flush
    float* outp = out + (size_t)cur_i * M_TOT * F_DIM;
#pragma unroll
    for (int mm = 0; mm < M_TOT; ++mm)
#pragma unroll
      for (int q = 0; q < 4; ++q)
        unsafeAtomicAdd(&outp[mm * F_DIM + q * 32 + lane], accz[mm * 4 + q]);
  }
}

// ---------------------------------------------------------------------------
extern "C" void kernel_launch(void* const* d_in, const int* in_sizes, int n_in,
                              void* d_out, int out_size, void* d_ws, size_t ws_size,
                              hipStream_t stream) {
  const float* x     = (const float*)d_in[0];
  const float* rbf   = (const float*)d_in[1];
  const float* phi   = (const float*)d_in[2];
  const float* W_Q   = (const float*)d_in[3];
  const float* W_K   = (const float*)d_in[4];
  const float* mW1   = (const float*)d_in[5];
  const float* mb1   = (const float*)d_in[6];
  const float* mW2   = (const float*)d_in[7];
  const float* mb2   = (const float*)d_in[8];
  const int*   idx_i = (const int*)d_in[9];
  const int*   idx_j = (const int*)d_in[10];
  const float* mask  = (const float*)d_in[11];

  float* out = (float*)d_out;
  float* Qws = (float*)d_ws;                               // 12.6 MB
  float* Kws = Qws + (size_t)N_ATOMS * N_DEG * F_DIM;      // 12.6 MB

  const int n4 = out_size / 4;
  zero_kernel<<<(n4 + 255) / 256, 256, 0, stream>>>((float4*)out, n4);
  qk_project_kernel<<<N_ATOMS / 16, 256, 0, stream>>>(x, W_Q, W_K, Qws, Kws);
  pair_kernel<<<N_PAIR / 128, 256, 0, stream>>>(rbf, phi, mW1, mb1, mW2, mb2,
                                                idx_i, idx_j, mask, Qws, Kws, x, out);
}